// AGNN_14491219656880
// MI455X (gfx1250) — compile-verified
//
#include <hip/hip_runtime.h>
#include <hip/hip_bf16.h>

#define IN_DIM 58
#define HID    100
#define RPB    128        // rows per block in GEMM
#define KP     60         // padded K (15 steps of 4)
#define NP     112        // padded N (7 col tiles of 16)

typedef float v2f __attribute__((ext_vector_type(2)));
typedef float v8f __attribute__((ext_vector_type(8)));

// ---------------------------------------------------------------------------
// h = relu(x @ W1^T + b1) via V_WMMA_F32_16X16X4_F32, LDS-staged operands.
// Block: 256 thr / 8 waves; stages aLDS[128][60] (padded x rows) and
// wLDS[60][112] (padded W1^T). Each wave computes one 16-row block across all
// 7 column tiles; inner K loop is branch-free (EXEC all-1s at every WMMA).
// A (16x4 f32): lanes 0-15 -> M=lane, {K+0,K+1}; lanes 16-31 -> M=lane-16, {K+2,K+3}.
// C/D (16x16 f32): VGPR r: lanes 0-15 M=r, N=lane; lanes 16-31 M=r+8, N=lane-16.
// ---------------------------------------------------------------------------
__global__ __launch_bounds__(256) void gemm1_relu_kernel(
    const float* __restrict__ x, const float* __restrict__ w1,
    const float* __restrict__ b1, float* __restrict__ h, int nNodes) {
  __shared__ float aLDS[RPB * KP];   // 30720 B
  __shared__ float wLDS[KP * NP];    // 26880 B

  const int tid  = threadIdx.x;
  const int wave = tid >> 5;
  const int lane = tid & 31;
  const int rowBase = blockIdx.x * RPB;

  // Stage padded W1^T: wLDS[k*NP + n] = w1[n][k], zero outside 58x100.
  for (int i = tid; i < KP * NP; i += 256) {
    const int k = i / NP, n = i % NP;
    wLDS[i] = (k < IN_DIM && n < HID) ? w1[n * IN_DIM + k] : 0.f;
  }
  // Stage padded x rows: aLDS[r*KP + k]; clamp OOB rows (never stored back).
  for (int i = tid; i < RPB * KP; i += 256) {
    const int r = i / KP, k = i % KP;
    int gr = rowBase + r; if (gr >= nNodes) gr = nNodes - 1;
    aLDS[i] = (k < IN_DIM) ? x[(long)gr * IN_DIM + k] : 0.f;
  }
  __syncthreads();

  const int m16   = lane & 15;
  const int khalf = (lane >> 4) * 2;            // 0 | 2
  const int mBase = (lane >> 4) * 8;
  const float* __restrict__ aRow = &aLDS[(wave * 16 + m16) * KP];

  for (int colBlk = 0; colBlk < 7; ++colBlk) {
    const int col0 = colBlk * 16;
    v8f acc = {};
    #pragma unroll
    for (int k = 0; k < KP; k += 4) {
      const int k0 = k + khalf;
      v2f a; a.x = aRow[k0];                        // ds_load_b64 (8B aligned)
             a.y = aRow[k0 + 1];
      v2f b; b.x = wLDS[ k0      * NP + col0 + m16];
             b.y = wLDS[(k0 + 1) * NP + col0 + m16];
      acc = __builtin_amdgcn_wmma_f32_16x16x4_f32(
          false, a, false, b, (short)0, acc, false, false);
    }
    const int nIdx = col0 + m16;
    if (nIdx < HID) {
      const float bias = b1[nIdx];
      #pragma unroll
      for (int rr = 0; rr < 8; ++rr) {
        const int m = rowBase + wave * 16 + mBase + rr;
        if (m < nNodes) {
          float v = acc[rr] + bias;
          h[(long)m * HID + nIdx] = v > 0.f ? v : 0.f;
        }
      }
    }
  }
}

// ---------------------------------------------------------------------------
// Per-node 1/max(||h||,1e-12). One wave per node; coalesced 400B row read.
// ---------------------------------------------------------------------------
__global__ __launch_bounds__(256) void inv_norm_kernel(
    const float* __restrict__ h, float* __restrict__ inv, int nNodes) {
  const int node = blockIdx.x * 8 + (threadIdx.x >> 5);
  const int lane = threadIdx.x & 31;
  if (node >= nNodes) return;
  const float* row = h + (long)node * HID;
  float s = 0.f;
  for (int d = lane; d < HID; d += 32) { float v = row[d]; s += v * v; }
  #pragma unroll
  for (int off = 16; off > 0; off >>= 1) s += __shfl_xor(s, off, 32);
  if (lane == 0) inv[node] = 1.f / fmaxf(sqrtf(s), 1e-12f);
}

__global__ __launch_bounds__(256) void init_softmax_kernel(
    float* __restrict__ m, float* __restrict__ den, int nNodes) {
  const int i = blockIdx.x * blockDim.x + threadIdx.x;
  if (i < nNodes) { m[i] = -__builtin_inff(); den[i] = 0.f; }
}

__global__ __launch_bounds__(256) void zero_kernel(float* __restrict__ p, long n) {
  const long i = (long)blockIdx.x * blockDim.x + threadIdx.x;
  if (i < n) p[i] = 0.f;
}

// Float atomic max via sign-split int/uint atomics (monotone IEEE bit trick).
__device__ __forceinline__ void atomicMaxFloat(float* addr, float val) {
  if (val >= 0.f) atomicMax((int*)addr, __float_as_int(val));
  else            atomicMin((unsigned int*)addr, __float_as_uint(val));
}

// ---------------------------------------------------------------------------
// Per-edge attention logit + fused segment-max. One wave per edge.
// alpha[e] = beta * dot(h[s],h[d]) * inv[s] * inv[d]
// ---------------------------------------------------------------------------
__global__ __launch_bounds__(256) void edge_alpha_kernel(
    const float* __restrict__ h, const float* __restrict__ inv,
    const int* __restrict__ srcA, const int* __restrict__ dstA,
    int nEdges, int total, const float* __restrict__ betaPtr,
    float* __restrict__ alpha, float* __restrict__ mOut) {
  const int e = blockIdx.x * 8 + (threadIdx.x >> 5);
  const int lane = threadIdx.x & 31;
  if (e >= total) return;
  const int s = (e < nEdges) ? srcA[e] : (e - nEdges);
  const int d = (e < nEdges) ? dstA[e] : (e - nEdges);
  const float* hs = h + (long)s * HID;
  const float* hd = h + (long)d * HID;
  float acc = 0.f;
  for (int i = lane; i < HID; i += 32) acc += hs[i] * hd[i];
  #pragma unroll
  for (int off = 16; off > 0; off >>= 1) acc += __shfl_xor(acc, off, 32);
  if (lane == 0) {
    const float beta = betaPtr ? betaPtr[0] : 1.0f;
    const float a = beta * acc * inv[s] * inv[d];
    alpha[e] = a;
    atomicMaxFloat(&mOut[d], a);
  }
}

// Per-edge exp + denominator accumulation (thread per edge).
__global__ __launch_bounds__(256) void edge_exp_kernel(
    const int* __restrict__ dstA, int nEdges, int total,
    const float* __restrict__ m, float* __restrict__ alpha,
    float* __restrict__ den) {
  const int e = blockIdx.x * blockDim.x + threadIdx.x;
  if (e >= total) return;
  const int d = (e < nEdges) ? dstA[e] : (e - nEdges);
  const float ev = __expf(alpha[e] - m[d]);
  alpha[e] = ev;
  atomicAdd(&den[d], ev);
}

// Attention-weighted scatter-add of source rows. One wave per edge.
__global__ __launch_bounds__(256) void edge_agg_kernel(
    const float* __restrict__ h,
    const int* __restrict__ srcA, const int* __restrict__ dstA,
    int nEdges, int total, const float* __restrict__ alpha,
    const float* __restrict__ den, float* __restrict__ out) {
  const int e = blockIdx.x * 8 + (threadIdx.x >> 5);
  const int lane = threadIdx.x & 31;
  if (e >= total) return;
  const int s = (e < nEdges) ? srcA[e] : (e - nEdges);
  const int d = (e < nEdges) ? dstA[e] : (e - nEdges);
  const float w = alpha[e] / (den[d] + 1e-16f);
  const float* hs = h + (long)s * HID;
  float* od = out + (long)d * HID;
  for (int i = lane; i < HID; i += 32) atomicAdd(&od[i], hs[i] * w);
}

// Final 100 -> 1 projection. One wave per node.
__global__ __launch_bounds__(256) void final_linear_kernel(
    const float* __restrict__ h, const float* __restrict__ w2,
    const float* __restrict__ b2, float* __restrict__ out, int nNodes) {
  const int node = blockIdx.x * 8 + (threadIdx.x >> 5);
  const int lane = threadIdx.x & 31;
  if (node >= nNodes) return;
  const float* row = h + (long)node * HID;
  float acc = 0.f;
  for (int i = lane; i < HID; i += 32) acc += row[i] * w2[i];
  #pragma unroll
  for (int off = 16; off > 0; off >>= 1) acc += __shfl_xor(acc, off, 32);
  if (lane == 0) out[node] = acc + b2[0];
}

// ---------------------------------------------------------------------------
// Host orchestration (all on `stream`, graph-capture safe).
// ---------------------------------------------------------------------------
static inline void run_prop(const float* hIn, float* hOut, const float* betaPtr,
                            const int* srcA, const int* dstA, int nEdges, int total,
                            float* inv, float* m, float* den, float* alpha,
                            int nNodes, hipStream_t stream) {
  const int waveBlocksN = (nNodes + 7) / 8;
  const int waveBlocksE = (total + 7) / 8;
  inv_norm_kernel<<<waveBlocksN, 256, 0, stream>>>(hIn, inv, nNodes);
  init_softmax_kernel<<<(nNodes + 255) / 256, 256, 0, stream>>>(m, den, nNodes);
  edge_alpha_kernel<<<waveBlocksE, 256, 0, stream>>>(hIn, inv, srcA, dstA, nEdges,
                                                     total, betaPtr, alpha, m);
  edge_exp_kernel<<<(total + 255) / 256, 256, 0, stream>>>(dstA, nEdges, total, m,
                                                           alpha, den);
  const long hElems = (long)nNodes * HID;
  zero_kernel<<<(int)((hElems + 255) / 256), 256, 0, stream>>>(hOut, hElems);
  edge_agg_kernel<<<waveBlocksE, 256, 0, stream>>>(hIn, srcA, dstA, nEdges, total,
                                                   alpha, den, hOut);
}

extern "C" void kernel_launch(void* const* d_in, const int* in_sizes, int n_in,
                              void* d_out, int out_size, void* d_ws, size_t ws_size,
                              hipStream_t stream) {
  const float* x     = (const float*)d_in[0];
  const int*   eidx  = (const int*)d_in[1];   // [2, E] int32 flat
  const float* w1    = (const float*)d_in[2];
  const float* b1    = (const float*)d_in[3];
  const float* beta2 = (const float*)d_in[4];
  const float* w2    = (const float*)d_in[5];
  const float* b2    = (const float*)d_in[6];
  float* out = (float*)d_out;

  const int nNodes = in_sizes[0] / IN_DIM;     // 50000
  const int nEdges = in_sizes[1] / 2;          // 800000
  const int total  = nEdges + nNodes;          // + self loops
  const int* srcA = eidx;
  const int* dstA = eidx + nEdges;

  // Workspace layout (floats).
  const long hElems = (long)nNodes * HID;
  float* h1    = (float*)d_ws;
  float* h2    = h1 + hElems;
  float* inv   = h2 + hElems;
  float* m     = inv + nNodes;
  float* den   = m + nNodes;
  float* alpha = den + nNodes;                 // total floats

  // 1) h1 = relu(x @ W1^T + b1) via WMMA FP32, LDS-staged.
  gemm1_relu_kernel<<<(nNodes + RPB - 1) / RPB, 256, 0, stream>>>(x, w1, b1, h1,
                                                                  nNodes);

  // 2) prop1 (beta fixed at 1): h1 -> h2
  run_prop(h1, h2, /*betaPtr=*/nullptr, srcA, dstA, nEdges, total,
           inv, m, den, alpha, nNodes, stream);

  // 3) prop2 (learnable beta): h2 -> h1
  run_prop(h2, h1, beta2, srcA, dstA, nEdges, total,
           inv, m, den, alpha, nNodes, stream);

  // 4) out = h1 @ W2^T + b2
  final_linear_kernel<<<(nNodes + 7) / 8, 256, 0, stream>>>(h1, w2, b2, out, nNodes);
}